// Attention_78889959293500
// MI455X (gfx1250) — compile-verified
//
#include <hip/hip_runtime.h>
#include <hip/hip_bf16.h>

typedef float v2f __attribute__((ext_vector_type(2)));
typedef float v4f __attribute__((ext_vector_type(4)));
typedef float v8f __attribute__((ext_vector_type(8)));

namespace {
constexpr int kB   = 32;
constexpr int kD   = 256;
constexpr int kT   = 32;
constexpr int kDh  = 128;
constexpr int kN   = 128 * 128;  // 16384 spatial positions
constexpr int kNT  = 128;        // n-tile per workgroup
constexpr int kSS  = kT + 2;     // 34: even (b64-aligned pairs) + conflict-free
constexpr int kVTS = kDh + 4;    // 132: transposed-V stride, even + conflict-free
}

// ---------------------------------------------------------------------------
// Kernel 1: values[b][k][t] = sum_d W[k][d] * wf[b][d][t] + bias[k]
// One workgroup per batch; wf[b] (32 KB) staged in LDS. W row is wave-uniform
// (scalar loads); LDS reads are lane==t, conflict-free.
// ---------------------------------------------------------------------------
__global__ __launch_bounds__(256) void values_kernel(
    const float* __restrict__ wf, const float* __restrict__ W,
    const float* __restrict__ bias, float* __restrict__ values) {
  __shared__ __align__(16) float swf[kD * kT];  // 32 KB
  const int b = blockIdx.x;
  const int tid = threadIdx.x;
  const v4f* src = (const v4f*)(wf + (size_t)b * kD * kT);
  v4f* dst4 = (v4f*)swf;
#pragma unroll
  for (int i = 0; i < (kD * kT / 4) / 256; ++i)
    dst4[tid + i * 256] = src[tid + i * 256];
  __syncthreads();
#pragma unroll
  for (int i = 0; i < (kDh * kT) / 256; ++i) {
    const int o = tid + i * 256;
    const int k = o >> 5;   // Dh index (uniform across a wave)
    const int t = o & 31;   // T index == lane
    float acc = bias[k];
    const float* wrow = W + k * kD;
#pragma unroll 4
    for (int d = 0; d < kD; ++d)
      acc = fmaf(wrow[d], swf[d * kT + t], acc);
    values[((b * kDh) + k) * kT + t] = acc;
  }
}

// ---------------------------------------------------------------------------
// Kernel 2: fused  S = Q^T V  ->  masked softmax  ->  attn = V P^T
// grid (N/128, B), 256 threads (8 wave32). fp32 WMMA 16x16x4 throughout.
// V kept transposed in LDS so every hot-loop B-fragment is one ds_load_b64.
// ---------------------------------------------------------------------------
__global__ __launch_bounds__(256) void attn_kernel(
    const float* __restrict__ image, const int* __restrict__ mask,
    const float* __restrict__ values, float* __restrict__ attn_out,
    float* __restrict__ coeff_out) {
  __shared__ __align__(16) float sVT[kT * kVTS];   // V^T: 32 x 132 = 16.9 KB
  __shared__ __align__(16) float sS[kNT * kSS];    // S/P tile: 128 x 34 = 17.4 KB
  __shared__ float sBias[kT];                      // 0 or -inf per t

  const int b   = blockIdx.y;
  const int n0  = blockIdx.x * kNT;
  const int tid = threadIdx.x;
  const int wv  = tid >> 5;        // wave id 0..7 (wave32)
  const int lane = tid & 31;
  const int hi  = lane >> 4;       // half-wave select
  const int l   = lane & 15;

  // ---- stage 0: load V[b] (row-major in ws) -> transposed LDS, + mask bias
  {
    const v4f* src = (const v4f*)(values + (size_t)b * kDh * kT);
#pragma unroll
    for (int i = 0; i < (kDh * kT / 4) / 256; ++i) {
      const int idx = tid + i * 256;
      const v4f f = src[idx];
      const int o = idx * 4;
      const int k = o >> 5;        // Dh row
      const int t = o & 31;        // T col (multiple of 4)
      sVT[(t + 0) * kVTS + k] = f.x;
      sVT[(t + 1) * kVTS + k] = f.y;
      sVT[(t + 2) * kVTS + k] = f.z;
      sVT[(t + 3) * kVTS + k] = f.w;
    }
    if (tid < kT)
      sBias[tid] = (mask[b * kT + tid] != 0) ? -__builtin_inff() : 0.0f;
  }
  __syncthreads();

  // ---- stage 1: S[n,t] = sum_k Q[k,n] V[k,t]
  // A(16x4 f32): lane l holds M=l; low half K={kk,kk+1}, high half K={kk+2,kk+3}
  // B(4x16 f32): lane l holds N=l; same half-wave K split -> v2f from sVT.
  const float* Qb = image + (size_t)b * kDh * kN;
  const int nrow = n0 + wv * 16 + l;
  v8f acc0 = {};  // t = 0..15
  v8f acc1 = {};  // t = 16..31
  for (int kk = 0; kk < kDh; kk += 4) {
    const int k0 = kk + 2 * hi;
    v2f a;
    a.x = __builtin_nontemporal_load(Qb + (size_t)k0 * kN + nrow);
    a.y = __builtin_nontemporal_load(Qb + (size_t)(k0 + 1) * kN + nrow);
    const v2f b0 = *(const v2f*)(sVT + l * kVTS + k0);          // t = l
    const v2f b1 = *(const v2f*)(sVT + (16 + l) * kVTS + k0);   // t = 16+l
    acc0 = __builtin_amdgcn_wmma_f32_16x16x4_f32(false, a, false, b0,
                                                 (short)0, acc0, false, false);
    acc1 = __builtin_amdgcn_wmma_f32_16x16x4_f32(false, a, false, b1,
                                                 (short)0, acc1, false, false);
  }
  // spill S stripe to LDS (C/D layout: vgpr j -> M = j + 8*hi, N = l)
#pragma unroll
  for (int j = 0; j < 8; ++j) {
    const int row = wv * 16 + j + 8 * hi;
    sS[row * kSS + l]      = acc0[j];
    sS[row * kSS + 16 + l] = acc1[j];
  }
  __syncthreads();

  // ---- masked softmax over t (one thread per n-row), stream coefficients out
  if (tid < kNT) {
    float s[kT];
    float m = -__builtin_inff();
#pragma unroll
    for (int t = 0; t < kT; ++t) {
      s[t] = sS[tid * kSS + t] + sBias[t];
      m = fmaxf(m, s[t]);
    }
    float sum = 0.0f;
#pragma unroll
    for (int t = 0; t < kT; ++t) { s[t] = __expf(s[t] - m); sum += s[t]; }
    const float inv = 1.0f / sum;
#pragma unroll
    for (int t = 0; t < kT; ++t) {
      const float p = s[t] * inv;
      sS[tid * kSS + t] = p;
      s[t] = p;
    }
    v4f* crow = (v4f*)(coeff_out + ((size_t)b * kN + (n0 + tid)) * kT);
#pragma unroll
    for (int t4 = 0; t4 < kT / 4; ++t4) {
      v4f v;
      v.x = s[4 * t4 + 0];
      v.y = s[4 * t4 + 1];
      v.z = s[4 * t4 + 2];
      v.w = s[4 * t4 + 3];
      __builtin_nontemporal_store(v, crow + t4);
    }
  }
  __syncthreads();

  // ---- stage 2: attn[k,n] = sum_t V[k,t] P[n,t]
  // wave wv owns Dh rows 16*wv..16*wv+15; loop over 8 n-subtiles.
  v2f aV[kT / 4];
#pragma unroll
  for (int q = 0; q < kT / 4; ++q) {
    const int tt = 4 * q + 2 * hi;
    aV[q].x = sVT[(tt + 0) * kVTS + wv * 16 + l];
    aV[q].y = sVT[(tt + 1) * kVTS + wv * 16 + l];
  }
  float* outb = attn_out + (size_t)b * kDh * kN;
  for (int nc = 0; nc < kNT / 16; ++nc) {
    v8f acc = {};
#pragma unroll
    for (int q = 0; q < kT / 4; ++q) {
      const v2f bf = *(const v2f*)(sS + (nc * 16 + l) * kSS + 4 * q + 2 * hi);
      acc = __builtin_amdgcn_wmma_f32_16x16x4_f32(false, aV[q], false, bf,
                                                  (short)0, acc, false, false);
    }
#pragma unroll
    for (int j = 0; j < 8; ++j) {
      const int krow = wv * 16 + j + 8 * hi;
      __builtin_nontemporal_store(
          acc[j], outb + (size_t)krow * kN + (n0 + nc * 16 + l));
    }
  }
}

// ---------------------------------------------------------------------------
extern "C" void kernel_launch(void* const* d_in, const int* in_sizes, int n_in,
                              void* d_out, int out_size, void* d_ws, size_t ws_size,
                              hipStream_t stream) {
  const float* wf    = (const float*)d_in[0];  // [B, D, T]
  const float* image = (const float*)d_in[1];  // [B, Dh, H, W]
  const int*   msk   = (const int*)d_in[2];    // [B, T]
  const float* W     = (const float*)d_in[3];  // [Dh, D]
  const float* bias  = (const float*)d_in[4];  // [Dh]
  float* out = (float*)d_out;                  // attn [B,Dh,N] ++ coeffs [B,N,T]
  float* vals = (float*)d_ws;                  // [B, Dh, T] scratch (512 KB)

  values_kernel<<<dim3(kB), 256, 0, stream>>>(wf, W, bias, vals);
  attn_kernel<<<dim3(kN / kNT, kB), 256, 0, stream>>>(
      image, msk, vals, out, out + (size_t)kB * kDh * kN);
}